// TopKGate_18425409700090
// MI455X (gfx1250) — compile-verified
//
#include <hip/hip_runtime.h>
#include <hip/hip_bf16.h>
#include <math.h>

typedef __attribute__((ext_vector_type(2))) float v2f;
typedef __attribute__((ext_vector_type(8))) float v8f;

#define INPUT_DIM 2048
#define N_EXPERTS 64
#define N_TOKENS  16384

// Block: 128 threads = 4 waves. One 16-token tile per block.
// Wave w computes scores for experts [16w, 16w+16) via V_WMMA_F32_16X16X4_F32.
__global__ __launch_bounds__(128) void topk_gate_kernel(
    const float* __restrict__ x,   // [N_TOKENS, INPUT_DIM]
    const float* __restrict__ W,   // [N_EXPERTS, INPUT_DIM]
    const float* __restrict__ b,   // [N_EXPERTS]
    float* __restrict__ out)       // [N_TOKENS, N_EXPERTS]
{
    __shared__ float sScores[16 * 65];   // padded rows: no bank conflicts
    __shared__ float sW1[16], sW2[16];
    __shared__ int   sI1[16], sI2[16];

    const int tid  = threadIdx.x;
    const int lane = tid & 31;
    const int wv   = tid >> 5;          // expert tile 0..3
    const int half = lane >> 4;         // 0: K0/K1 or M0..7, 1: K2/K3 or M8..15
    const int idx  = lane & 15;         // A: row m; B: expert-in-tile n
    const int rowBase = blockIdx.x * 16;

    // A fragment (32-bit A 16x4): lane<16 holds (M=idx, K=k0..k0+1),
    // lane>=16 holds (M=idx, K=k0+2..k0+3) -> contiguous 8B per lane.
    const float* aPtr = x + (size_t)(rowBase + idx) * INPUT_DIM + 2 * half;
    // B fragment (4x16, K rows striped: VGPR v holds K=v + 2*half, N=idx lanes)
    const float* bPtr = W + (size_t)(wv * 16 + idx) * INPUT_DIM + 2 * half;

    v8f c = {};
    #pragma unroll 8
    for (int k0 = 0; k0 < INPUT_DIM; k0 += 4) {
        v2f a  = *(const v2f*)(aPtr + k0);
        v2f bb = *(const v2f*)(bPtr + k0);
        // D = A(16x4) * B(4x16) + C, full f32
        c = __builtin_amdgcn_wmma_f32_16x16x4_f32(
                /*neg_a=*/false, a, /*neg_b=*/false, bb,
                /*c_mod=*/(short)0, c, /*reuse_a=*/false, /*reuse_b=*/false);
    }

    // bias: expert e = 16*wv + idx is fixed per lane across all 8 C rows
    const float bias = b[wv * 16 + idx];
    #pragma unroll
    for (int v = 0; v < 8; ++v) {
        const int m = v + 8 * half;                    // C row layout
        sScores[m * 65 + wv * 16 + idx] = c[v] + bias; // conflict-free: e contiguous
    }
    __syncthreads();

    // Per-token top-2 + full softmax denominator + renormalization.
    if (tid < 16) {
        const int m = tid;
        const float* row = &sScores[m * 65];
        float m1 = -INFINITY, m2 = -INFINITY;
        int   i1 = 0, i2 = 0;
        #pragma unroll 4
        for (int e = 0; e < 64; ++e) {
            const float s = row[e];
            if (s > m1)      { m2 = m1; i2 = i1; m1 = s; i1 = e; }
            else if (s > m2) { m2 = s;  i2 = e; }
        }
        float Z = 0.f;
        #pragma unroll 4
        for (int e = 0; e < 64; ++e) Z += expf(row[e] - m1);
        const float p1 = 1.0f / Z;             // exp(m1-m1)/Z
        const float p2 = expf(m2 - m1) / Z;
        const float denom = p1 + p2 + 1e-8f;
        sW1[m] = p1 / denom;
        sW2[m] = p2 / denom;
        sI1[m] = i1;
        sI2[m] = i2;
    }
    __syncthreads();

    // Emit dense 16x64 tile: 1024 floats, 128 threads x 2 float4 (coalesced).
    float4* out4 = (float4*)(out + (size_t)rowBase * N_EXPERTS);
    #pragma unroll
    for (int j = 0; j < 2; ++j) {
        const int q4 = tid * 2 + j;        // float4 index in tile
        const int p  = q4 * 4;             // float index
        const int m  = p >> 6;             // token in tile
        const int e0 = p & 63;             // first expert of this float4
        const int i1 = sI1[m], i2 = sI2[m];
        const float w1 = sW1[m], w2 = sW2[m];
        float v4[4];
        #pragma unroll
        for (int q = 0; q < 4; ++q) {
            const int e = e0 + q;
            v4[q] = (e == i1) ? w1 : ((e == i2) ? w2 : 0.0f);
        }
        out4[q4] = make_float4(v4[0], v4[1], v4[2], v4[3]);
    }
}

extern "C" void kernel_launch(void* const* d_in, const int* in_sizes, int n_in,
                              void* d_out, int out_size, void* d_ws, size_t ws_size,
                              hipStream_t stream) {
    const float* x = (const float*)d_in[0];
    const float* W = (const float*)d_in[1];
    const float* b = (const float*)d_in[2];
    float* out = (float*)d_out;

    dim3 grid(N_TOKENS / 16);
    dim3 block(128);
    topk_gate_kernel<<<grid, block, 0, stream>>>(x, W, b, out);
}